// KNNReducer_71227737637484
// MI455X (gfx1250) — compile-verified
//
#include <hip/hip_runtime.h>
#include <hip/hip_bf16.h>
#include <stdint.h>

// ---------------------------------------------------------------------------
// KNN gather-mean on MI455X (gfx1250)
//
// Memory-bound gather: x (67MB) fits in the 192MB L2, so the 8x gather
// amplification is L2-resident; HBM traffic ~134MB -> ~6us floor at 23.3TB/s.
// CDNA5 Tensor Data Mover in GATHER MODE (D# gather_mode=1, 32-bit indices,
// 8 rows == K) DMAs the 8 neighbor rows per output position straight into
// LDS. Each wave queues FOUR gathers up-front (TDM ops from one wave
// complete in order), then drains them with descending s_wait_tensorcnt,
// overlapping DMA latency with the FP32 reductions.
// ---------------------------------------------------------------------------

typedef unsigned int u32x4 __attribute__((ext_vector_type(4)));
typedef int          i32x8 __attribute__((ext_vector_type(8)));
typedef int          i32x4 __attribute__((ext_vector_type(4)));

#define N_SPATIAL 4096
#define FDIM      64
#define KNN       8
#define WAVES_PER_BLOCK 8
#define ROWS_PER_WAVE   4              // == pipeline depth (one buffer per row)
#define TILE_FLOATS (KNN * FDIM)       // 512 floats = 2KB gathered tile
#define TILE_BYTES  (TILE_FLOATS * 4)

// Issue one TDM gather: 8 rows of 64 f32, selected by idx8[0..7], into LDS.
__device__ __forceinline__ void tdm_gather8(const float* slice_base,
                                            const int*   idx8,
                                            unsigned     lds_byte_off)
{
    uint64_t ga = (uint64_t)(uintptr_t)slice_base;

    // D# group 0: count=1 | gather_index_size=32b (bit30) | gather_mode (bit31)
    u32x4 g0;
    g0[0] = 0xC0000001u;
    g0[1] = lds_byte_off;                                   // lds_addr
    g0[2] = (unsigned)ga;                                   // global_addr[31:0]
    g0[3] = (unsigned)((ga >> 32) & 0x1FFFFFFu)             // global_addr[56:32]
          | 0x80000000u;                                    // type=2 ("image")

    // D# group 1
    i32x8 g1;
    g1[0] = 0x00020000;                 // workgroup_mask=0, data_size=2 (4B)
    g1[1] = (FDIM << 16);               // tensor_dim0 = 64 (bits 79:48)
    g1[2] = (N_SPATIAL & 0xFFFF) << 16; // tensor_dim1 = 4096 (bits 111:80)
    g1[3] = (FDIM << 16);               // tile_dim0 = 64 (bits 127:112)
    g1[4] = KNN;                        // tile_dim1 = 8 valid gather indices
    g1[5] = FDIM;                       // tensor_dim0_stride = 64 elements
    g1[6] = 0;
    g1[7] = 0;

    // D# groups 2/3: 32-bit row indices, 4 per group — knn row maps directly.
    i32x4 g2, g3;
    g2[0] = idx8[0]; g2[1] = idx8[1]; g2[2] = idx8[2]; g2[3] = idx8[3];
    g3[0] = idx8[4]; g3[1] = idx8[5]; g3[2] = idx8[6]; g3[3] = idx8[7];

    // Trailing operand of this toolchain's 6-arg builtin: unused group -> 0.
    i32x8 gz = {0, 0, 0, 0, 0, 0, 0, 0};

    __builtin_amdgcn_tensor_load_to_lds(g0, g1, g2, g3, gz, /*cpol=*/0);
}

// Reduce one 8x64 gathered tile to its 64-float mean and store it.
__device__ __forceinline__ void reduce_store(const float* tile,
                                             float*       out_row,
                                             int          lane)
{
    // Lane owns floats [2*lane, 2*lane+1] of each row.
    // 32 lanes * 8B = 256B per row -> all 64 LDS banks, conflict-free.
    const float2* buf = (const float2*)tile;
    float2 acc; acc.x = 0.0f; acc.y = 0.0f;
    #pragma unroll
    for (int k = 0; k < KNN; ++k) {
        float2 v = buf[k * (FDIM / 2) + lane];
        acc.x += v.x;
        acc.y += v.y;
    }
    acc.x *= 0.125f;
    acc.y *= 0.125f;
    ((float2*)out_row)[lane] = acc;    // coalesced 256B store per wave
}

__global__ __launch_bounds__(256)
void knn_mean_tdm(const float* __restrict__ x,
                  const int*   __restrict__ knn,
                  float*       __restrict__ out)
{
    // 8 waves * 4 buffers * 2KB = 64KB LDS per block (WGP pool is 320KB).
    __shared__ float lds[WAVES_PER_BLOCK * ROWS_PER_WAVE * TILE_FLOATS];

    const int lane = threadIdx.x & 31;
    const int wave = __builtin_amdgcn_readfirstlane((int)threadIdx.x >> 5);

    const long long row0 =
        ((long long)blockIdx.x * WAVES_PER_BLOCK + wave) * ROWS_PER_WAVE;

    float*   wbuf  = &lds[wave * ROWS_PER_WAVE * TILE_FLOATS];
    unsigned lbase = (unsigned)(uintptr_t)wbuf;   // low 32 bits = LDS offset

    // ---- Issue all four gathers back-to-back (4-deep TDM queue) ----------
    #pragma unroll
    for (int j = 0; j < ROWS_PER_WAVE; ++j) {
        long long r = row0 + j;
        int bt = (int)(r >> 12);
        int n  = (int)(r & (N_SPATIAL - 1));
        tdm_gather8(x + (long long)bt * N_SPATIAL * FDIM, knn + n * KNN,
                    lbase + (unsigned)(j * TILE_BYTES));
    }

    // ---- Drain in order: TDM ops from one wave complete in order ---------
    __builtin_amdgcn_s_wait_tensorcnt(3);
    reduce_store(&wbuf[0 * TILE_FLOATS], out + (row0 + 0) * (long long)FDIM, lane);

    __builtin_amdgcn_s_wait_tensorcnt(2);
    reduce_store(&wbuf[1 * TILE_FLOATS], out + (row0 + 1) * (long long)FDIM, lane);

    __builtin_amdgcn_s_wait_tensorcnt(1);
    reduce_store(&wbuf[2 * TILE_FLOATS], out + (row0 + 2) * (long long)FDIM, lane);

    __builtin_amdgcn_s_wait_tensorcnt(0);
    reduce_store(&wbuf[3 * TILE_FLOATS], out + (row0 + 3) * (long long)FDIM, lane);
}

extern "C" void kernel_launch(void* const* d_in, const int* in_sizes, int n_in,
                              void* d_out, int out_size, void* d_ws, size_t ws_size,
                              hipStream_t stream)
{
    const float* x   = (const float*)d_in[0];   // [4,16,64,64,64] f32
    const int*   knn = (const int*)d_in[1];     // [4096, 8] i32
    float*       out = (float*)d_out;           // [4,16,64,64,64] f32

    // Total output rows: BT*N = 64*4096 = 262144.
    // Rows per block = 8 waves * 4 rows = 32 -> 8192 blocks.
    const int total_rows = 64 * N_SPATIAL;
    const int rows_per_block = WAVES_PER_BLOCK * ROWS_PER_WAVE;
    dim3 grid(total_rows / rows_per_block);
    dim3 block(256);
    hipLaunchKernelGGL(knn_mean_tdm, grid, block, 0, stream, x, knn, out);
}